// DynamicScatter3D_87763361726686
// MI455X (gfx1250) — compile-verified
//
#include <hip/hip_runtime.h>
#include <hip/hip_bf16.h>
#include <stdint.h>

// ---------------------------------------------------------------------------
// DynamicScatter3D (point->voxel mean) for MI455X / gfx1250
//   key space = 100^3 = 1e6  -> dense L2-resident tables (counts + rank)
//   pass 1: histogram (int atomics)
//   pass 2: exclusive scan of presence -> rank (== jnp.unique sorted order)
//   pass 3: double-buffered TDM-staged tiles -> pre-scaled f32 atomic scatter
// ---------------------------------------------------------------------------

#define VGRID   100
#define NKEYS   (VGRID * VGRID * VGRID)   // 1,000,000
#define NFEAT   64
#define SCAN_B  1024
#define BATCH   128                        // points per TDM tile (32 KB)
#define NB      4                          // batches per block (pipelined)

typedef unsigned int v4u __attribute__((ext_vector_type(4)));
typedef int          v8i __attribute__((ext_vector_type(8)));
typedef int          v4i __attribute__((ext_vector_type(4)));

__device__ __forceinline__ int voxel_key(const int* coors, int p) {
    return coors[3 * p] * (VGRID * VGRID) + coors[3 * p + 1] * VGRID + coors[3 * p + 2];
}

// Issue one TDM transfer: nElems f32 (1-row tile), global -> LDS.
__device__ __forceinline__ void tdm_issue(const float* gsrc, float* ldst, int nElems) {
    unsigned long long gaddr = (unsigned long long)(uintptr_t)gsrc;
    unsigned ldsOff = (unsigned)(uintptr_t)ldst;
    v4u g0;
    g0.x = 1u;                                            // count=1 (valid D#)
    g0.y = ldsOff;                                        // lds_addr
    g0.z = (unsigned)(gaddr & 0xFFFFFFFFull);             // global_addr[31:0]
    g0.w = (unsigned)((gaddr >> 32) & 0x1FFFFFFull)       // global_addr[56:32]
           | 0x80000000u;                                 // type=2 ("image")
    v8i g1;
    g1[0] = 0x00020000;                                   // data_size = 4B
    g1[1] = (nElems & 0xFFFF) << 16;                      // tensor_dim0 lo16
    g1[2] = (nElems >> 16) | (1 << 16);                   // dim0 hi16 | tensor_dim1=1
    g1[3] = (nElems & 0xFFFF) << 16;                      // tile_dim0
    g1[4] = 1;                                            // tile_dim1 = 1
    g1[5] = nElems;                                       // tensor_dim0_stride lo32
    g1[6] = 0;
    g1[7] = 0;
    v4i g2 = {0, 0, 0, 0};
    v4i g3 = {0, 0, 0, 0};
    v8i g4 = {0, 0, 0, 0, 0, 0, 0, 0};                    // unused extra group
    __builtin_amdgcn_tensor_load_to_lds(g0, g1, g2, g3, g4, 0);
}

// ---- K0: init output (feats=0, coors=-1) and workspace, 128-bit stores ----
__global__ void ds3d_init(uint4* counts4, uint4* blockSums4,
                          float4* outF4, int4* outCoors4,
                          long long nfeat4, long long ncoor4,
                          long long nkeys4, long long nbs4) {
    long long stride = (long long)gridDim.x * blockDim.x;
    long long t0 = (long long)blockIdx.x * blockDim.x + threadIdx.x;
    float4 fz = make_float4(0.f, 0.f, 0.f, 0.f);
    int4   im = make_int4(-1, -1, -1, -1);
    uint4  uz = make_uint4(0u, 0u, 0u, 0u);
    for (long long i = t0; i < nfeat4; i += stride) outF4[i] = fz;
    for (long long i = t0; i < ncoor4; i += stride) outCoors4[i] = im;
    for (long long i = t0; i < nkeys4; i += stride) counts4[i] = uz;
    for (long long i = t0; i < nbs4;   i += stride) blockSums4[i] = uz;
}

// ---- K1: per-voxel point histogram ----------------------------------------
__global__ void ds3d_count(const int* __restrict__ coors, unsigned* counts, int n) {
    int p = blockIdx.x * blockDim.x + threadIdx.x;
    if (p < n) atomicAdd(&counts[voxel_key(coors, p)], 1u);
}

// ---- K2a: per-block presence sums -----------------------------------------
__global__ void ds3d_blocksum(const unsigned* __restrict__ counts, unsigned* blockSums) {
    __shared__ unsigned s[SCAN_B];
    int t = blockIdx.x * SCAN_B + threadIdx.x;
    s[threadIdx.x] = (t < NKEYS && counts[t] != 0u) ? 1u : 0u;
    __syncthreads();
    for (int off = SCAN_B / 2; off > 0; off >>= 1) {
        if ((int)threadIdx.x < off) s[threadIdx.x] += s[threadIdx.x + off];
        __syncthreads();
    }
    if (threadIdx.x == 0) blockSums[blockIdx.x] = s[0];
}

// ---- K2b: exclusive scan across block sums (single 1024-thread block) -----
__global__ void ds3d_scanblocks(unsigned* blockSums, int nblocks) {
    __shared__ unsigned s[SCAN_B];
    unsigned v = ((int)threadIdx.x < nblocks) ? blockSums[threadIdx.x] : 0u;
    s[threadIdx.x] = v;
    __syncthreads();
    for (int off = 1; off < SCAN_B; off <<= 1) {
        unsigned add = ((int)threadIdx.x >= off) ? s[threadIdx.x - off] : 0u;
        __syncthreads();
        s[threadIdx.x] += add;
        __syncthreads();
    }
    if ((int)threadIdx.x < nblocks) blockSums[threadIdx.x] = s[threadIdx.x] - v;
}

// ---- K2c: rank assignment (sorted-key compaction) + voxel coor decode -----
__global__ void ds3d_rank(const unsigned* __restrict__ counts,
                          const unsigned* __restrict__ blockSums,
                          int* rank, int* outCoors) {
    __shared__ unsigned s[SCAN_B];
    int t = blockIdx.x * SCAN_B + threadIdx.x;
    unsigned pres = (t < NKEYS && counts[t] != 0u) ? 1u : 0u;
    s[threadIdx.x] = pres;
    __syncthreads();
    for (int off = 1; off < SCAN_B; off <<= 1) {
        unsigned add = ((int)threadIdx.x >= off) ? s[threadIdx.x - off] : 0u;
        __syncthreads();
        s[threadIdx.x] += add;
        __syncthreads();
    }
    if (pres) {
        int r = (int)(blockSums[blockIdx.x] + s[threadIdx.x] - 1u);
        rank[t] = r;
        outCoors[3 * r + 0] = t / (VGRID * VGRID);
        outCoors[3 * r + 1] = (t / VGRID) % VGRID;
        outCoors[3 * r + 2] = t % VGRID;
    }
}

// ---- K3: double-buffered TDM pipeline + pre-scaled atomic scatter ---------
__global__ void ds3d_scatter(const float* __restrict__ feats,
                             const int* __restrict__ coors,
                             const unsigned* __restrict__ counts,
                             const int* __restrict__ rank,
                             float* outF, int nPoints) {
    __shared__ __align__(16) float sF[2][BATCH * NFEAT];   // 2 x 32 KB
    __shared__ int   sIdx[BATCH];
    __shared__ float sScale[BATCH];

    int totalBatches = (nPoints + BATCH - 1) / BATCH;
    int b0 = blockIdx.x * NB;
    if (b0 >= totalBatches) return;
    int nbLocal = totalBatches - b0;
    if (nbLocal > NB) nbLocal = NB;

    // Prime the pipeline: DMA batch b0 into buffer 0.
    if (threadIdx.x == 0) {
        int p0 = b0 * BATCH;
        int npts = nPoints - p0; if (npts > BATCH) npts = BATCH;
        tdm_issue(feats + (size_t)p0 * NFEAT, &sF[0][0], npts * NFEAT);
    }

    for (int i = 0; i < nbLocal; ++i) {
        int b = b0 + i;
        int p0 = b * BATCH;
        int npts = nPoints - p0; if (npts > BATCH) npts = BATCH;
        int nElems = npts * NFEAT;
        int par = i & 1;
        bool more = (i + 1 < nbLocal);

        // Issue next batch's DMA into the other buffer (safe: that buffer was
        // last read in iteration i-1, which ended with a barrier).
        if (threadIdx.x == 0 && more) {
            int p0n = (b + 1) * BATCH;
            int nptsN = nPoints - p0n; if (nptsN > BATCH) nptsN = BATCH;
            tdm_issue(feats + (size_t)p0n * NFEAT, &sF[par ^ 1][0], nptsN * NFEAT);
        }

        // Dependent gather chain (coors -> key -> rank/count) overlaps DMA.
        if ((int)threadIdx.x < npts) {
            int p = p0 + threadIdx.x;
            int key = voxel_key(coors, p);
            sIdx[threadIdx.x] = rank[key];
            unsigned c = counts[key];
            sScale[threadIdx.x] = 1.0f / (float)(c ? c : 1u);
        }

        // Tensor ops from one wave complete in order: <=1 outstanding means
        // batch i has landed in sF[par].
        if (threadIdx.x == 0) {
            if (more) __builtin_amdgcn_s_wait_tensorcnt(1);
            else      __builtin_amdgcn_s_wait_tensorcnt(0);
        }
        __syncthreads();

        // 64 consecutive lanes hit 64 consecutive words of one output row:
        // fully coalesced global_atomic_add_f32 (no-return), straight to L2.
        for (int e = threadIdx.x; e < nElems; e += blockDim.x) {
            int pl = e >> 6;
            int c  = e & (NFEAT - 1);
            atomicAdd(&outF[(size_t)sIdx[pl] * NFEAT + c], sF[par][e] * sScale[pl]);
        }
        __syncthreads();   // all reads of sF[par] done before it is re-filled
    }
}

// ---------------------------------------------------------------------------
extern "C" void kernel_launch(void* const* d_in, const int* in_sizes, int n_in,
                              void* d_out, int out_size, void* d_ws, size_t ws_size,
                              hipStream_t stream) {
    const float* feats = (const float*)d_in[0];
    const int*   coors = (const int*)d_in[1];
    int n = in_sizes[1] / 3;                       // 1,000,000 points

    unsigned* counts    = (unsigned*)d_ws;                                  // 4 MB
    int*      rank      = (int*)((char*)d_ws + (size_t)NKEYS * 4);          // 4 MB
    unsigned* blockSums = (unsigned*)((char*)d_ws + (size_t)NKEYS * 8);     // 4 KB

    float* outF     = (float*)d_out;                     // n x 64 f32
    int*   outCoors = (int*)(outF + (size_t)n * NFEAT);  // n x 3 i32 (bit view)

    int nScanBlocks = (NKEYS + SCAN_B - 1) / SCAN_B;     // 977

    // All element counts below are multiples of 4 (n = 1e6).
    ds3d_init<<<2048, 256, 0, stream>>>((uint4*)counts, (uint4*)blockSums,
                                        (float4*)outF, (int4*)outCoors,
                                        (long long)n * NFEAT / 4,
                                        (long long)n * 3 / 4,
                                        (long long)NKEYS / 4,
                                        (long long)SCAN_B / 4);
    ds3d_count<<<(n + 255) / 256, 256, 0, stream>>>(coors, counts, n);
    ds3d_blocksum<<<nScanBlocks, SCAN_B, 0, stream>>>(counts, blockSums);
    ds3d_scanblocks<<<1, SCAN_B, 0, stream>>>(blockSums, nScanBlocks);
    ds3d_rank<<<nScanBlocks, SCAN_B, 0, stream>>>(counts, blockSums, rank, outCoors);

    int totalBatches = (n + BATCH - 1) / BATCH;
    int nScatBlocks  = (totalBatches + NB - 1) / NB;
    ds3d_scatter<<<nScatBlocks, 256, 0, stream>>>(feats, coors, counts,
                                                  rank, outF, n);
}